// CorrelationHead_47124381172291
// MI455X (gfx1250) — compile-verified
//
#include <hip/hip_runtime.h>

typedef __attribute__((ext_vector_type(2))) float v2f;
typedef __attribute__((ext_vector_type(8))) float v8f;

#define NB   512      // batch
#define NC   128      // channels
#define NIJ  49       // 7*7 spatial positions
#define CHW  6272     // NC*NIJ
#define PP   21
#define FEAT 21609    // 21*21*49

// One workgroup per batch b.
//   Stage patch1[b] (128x49) and patch2[b] (128x49) into LDS with
//   GLOBAL_LOAD_ASYNC_TO_LDS_B128 (CDNA5 async DMA path, ASYNCcnt-tracked).
//   T[b] = patch1[b]^T * patch2[b]  (49x49, K=128) via V_WMMA_F32_16X16X4_F32,
//   16 waves, one 16x16 tile each (4x4 tile grid, edges clamped).
//   Head fused from accumulator registers: out[b,r] = bias[r] +
//     sum over (ij,uv) with matching parity of T * w[r, (p*21+q)*49 + ij].
__global__ __launch_bounds__(512)
void corr_head_kernel(const float* __restrict__ patch1,
                      const float* __restrict__ patch2,
                      const float* __restrict__ w,
                      const float* __restrict__ bias,
                      float* __restrict__ out)
{
    __shared__ float sA[CHW];        // patch1[b], layout [c][ij]
    __shared__ float sB[CHW];        // patch2[b], layout [c][uv]
    __shared__ float wred[16][4];    // per-wave partial sums for 4 outputs

    const int b   = blockIdx.x;
    const int tid = threadIdx.x;

    // ---- stage both patches into LDS via async global->LDS (no VGPR bounce) ----
    // LDS byte offset (group-segment relative) = low 32 bits of flat shared ptr.
    const unsigned sA_base = (unsigned)(size_t)(void*)sA;
    const unsigned sB_base = (unsigned)(size_t)(void*)sB;
    const float* gA = patch1 + (size_t)b * CHW;
    const float* gB = patch2 + (size_t)b * CHW;
    for (int idx = tid; idx < CHW / 4; idx += 512) {
        unsigned la = sA_base + (unsigned)idx * 16u;
        unsigned lb = sB_base + (unsigned)idx * 16u;
        const float* ga = gA + idx * 4;
        const float* gbp = gB + idx * 4;
        asm volatile("global_load_async_to_lds_b128 %0, %1, off"
                     :: "v"(la), "v"(ga) : "memory");
        asm volatile("global_load_async_to_lds_b128 %0, %1, off"
                     :: "v"(lb), "v"(gbp) : "memory");
    }
    // drain this wave's async transfers, then workgroup barrier => LDS coherent
    asm volatile("s_wait_asynccnt 0" ::: "memory");
    __syncthreads();

    // ---- per-wave 16x16 tile of T = A^T B via fp32 WMMA ----
    const int wave = tid >> 5;
    const int lane = tid & 31;
    const int half = lane >> 4;          // 0: K pair {k,k+1}, 1: {k+2,k+3}
    const int l16  = lane & 15;
    const int m0   = (wave >> 2) * 16;   // tile row origin (ij)
    const int n0   = (wave & 3) * 16;    // tile col origin (uv)
    // clamp edge tiles (rows/cols >=49 are discarded later; clamping only
    // keeps LDS addresses in range, and keeps EXEC uniform for WMMA)
    const int mi = (m0 + l16 < NIJ) ? (m0 + l16) : (NIJ - 1);
    const int ni = (n0 + l16 < NIJ) ? (n0 + l16) : (NIJ - 1);

    v8f acc = {};
    for (int k = 0; k < NC; k += 4) {
        const int kk = k + 2 * half;
        v2f av, bv;
        av.x = sA[kk * NIJ + mi];
        av.y = sA[(kk + 1) * NIJ + mi];
        bv.x = sB[kk * NIJ + ni];
        bv.y = sB[(kk + 1) * NIJ + ni];
        acc = __builtin_amdgcn_wmma_f32_16x16x4_f32(
            false, av, false, bv, (short)0, acc, false, false);
    }

    // ---- fused linear head straight from D registers ----
    // D layout: VGPR g -> row M = m0 + g + 8*half, col N = n0 + l16
    float p0 = 0.f, p1 = 0.f, p2 = 0.f, p3 = 0.f;
    #pragma unroll
    for (int g = 0; g < 8; ++g) {
        const int row = m0 + 8 * half + g;   // ij = i*7 + j
        const int col = n0 + l16;            // uv = u*7 + v
        if (row < NIJ && col < NIJ) {
            const int i = row / 7, j = row - i * 7;
            const int u = col / 7, v = col - u * 7;
            const int di = u - i, dj = v - j;
            if (((di | dj) & 1) == 0) {      // both displacements even
                const int p  = 10 + (di >> 1);
                const int q  = 10 + (dj >> 1);
                const int wi = (p * PP + q) * NIJ + row;
                const float tv = acc[g];
                p0 += tv * w[wi];
                p1 += tv * w[FEAT + wi];
                p2 += tv * w[2 * FEAT + wi];
                p3 += tv * w[3 * FEAT + wi];
            }
        }
    }

    // ---- deterministic reduction: wave shuffle, then fixed-order LDS sum ----
    #pragma unroll
    for (int off = 16; off > 0; off >>= 1) {
        p0 += __shfl_down(p0, off, 32);
        p1 += __shfl_down(p1, off, 32);
        p2 += __shfl_down(p2, off, 32);
        p3 += __shfl_down(p3, off, 32);
    }
    if (lane == 0) {
        wred[wave][0] = p0;
        wred[wave][1] = p1;
        wred[wave][2] = p2;
        wred[wave][3] = p3;
    }
    __syncthreads();

    if (tid < 4) {
        float s = bias[tid];
        #pragma unroll
        for (int wv = 0; wv < 16; ++wv) s += wred[wv][tid];
        out[b * 4 + tid] = s;
    }
}

extern "C" void kernel_launch(void* const* d_in, const int* in_sizes, int n_in,
                              void* d_out, int out_size, void* d_ws, size_t ws_size,
                              hipStream_t stream)
{
    const float* patch1 = (const float*)d_in[0];
    const float* patch2 = (const float*)d_in[1];
    const float* w_bbox = (const float*)d_in[2];
    const float* b_bbox = (const float*)d_in[3];
    float* out = (float*)d_out;

    corr_head_kernel<<<NB, 512, 0, stream>>>(patch1, patch2, w_bbox, b_bbox, out);
}